// Score_1202590843020
// MI455X (gfx1250) — compile-verified
//
#include <hip/hip_runtime.h>
#include <math.h>

// Problem constants (match reference)
#define BATCH 64
#define LSEQ  1024
#define DDIM  2048
#define NREG  20
#define RS    2052   // padded LDS row stride in floats (2048+4 -> bank-conflict-free)

#define THREADS 512          // 16 wave32 per workgroup (4 waves/SIMD on the WGP)
#define WAVES   (THREADS/32)
#define COLS_PER_WG (WAVES*16)         // 256 columns of L per workgroup
#define CHUNKS  (LSEQ/COLS_PER_WG)     // 4 chunks per batch

typedef __attribute__((ext_vector_type(2))) float v2f;
typedef __attribute__((ext_vector_type(4))) float v4f;
typedef __attribute__((ext_vector_type(8))) float v8f;

// One workgroup = one batch x 256 columns. 16 waves, each wave owns a
// 16-column strip. Regions (20x2048 f32, padded to 32 rows) live in LDS
// (256.5 KB) and are reused across all columns; words stream from HBM once.
__global__ __launch_bounds__(THREADS, 1)
void score_max_wmma_kernel(const float* __restrict__ regions,  // (B*20, D)
                           const float* __restrict__ words,    // (B, L, D)
                           float* __restrict__ out)            // (B, 1, L)
{
    extern __shared__ float lds[];

    const int tid  = threadIdx.x;
    const int lane = tid & 31;
    const int wave = tid >> 5;
    const int b        = blockIdx.x >> 2;           // batch index
    const int chunk    = blockIdx.x & (CHUNKS - 1); // which 256-col chunk of L
    const int col_base = chunk * COLS_PER_WG + wave * 16;

    // ---- Stage region matrix into LDS: rows 0..19 real, rows 20..31 zero ----
    // 32 rows x 2048 floats, moved as float4 (16384 vec4 / 512 threads).
    for (int i = tid; i < 32 * (DDIM / 4); i += THREADS) {
        const int e = i << 2;           // element index in 32x2048
        const int r = e >> 11;          // / 2048
        const int d = e & (DDIM - 1);   // % 2048
        v4f v = {0.f, 0.f, 0.f, 0.f};
        if (r < NREG)
            v = *reinterpret_cast<const v4f*>(regions + (((size_t)(b * NREG + r)) << 11) + d);
        *reinterpret_cast<v4f*>(&lds[r * RS + d]) = v;
    }
    __syncthreads();

    // ---- Per-lane operand addressing (V_WMMA_F32_16X16X4_F32 layouts) ----
    // A (16x4 MxK): lanes 0-15 hold {K,K+1} of row M=lane; lanes 16-31 hold
    // {K+2,K+3} of row M=lane-16.  B (4x16 KxN) mirrors with lane = column N.
    const int mrow = lane & 15;
    const int koff = (lane >> 4) << 1;                 // 0 or 2
    const float* gB  = words + ((size_t)b * LSEQ + (col_base + mrow)) * DDIM + koff;
    const float* lA0 = lds + mrow * RS + koff;          // tile 0: rows 0..15
    const float* lA1 = lA0 + 16 * RS;                   // tile 1: rows 16..31

    v8f acc0 = {0.f, 0.f, 0.f, 0.f, 0.f, 0.f, 0.f, 0.f};

    // Tile 1 covers M=16..31 but only rows 16..19 are real. In the C/D layout,
    // local rows 0..7 sit in lanes 0-15 (VGPR v -> M=v) and rows 8..15 in lanes
    // 16-31. Valid local rows are 0..3 in lanes 0-15; everything else starts at
    // -inf (zero A rows contribute 0 each step, so -inf survives the k-loop)
    // and is therefore ignored by the final max.
    const float NEG = -INFINITY;
    const bool lo = (lane < 16);
    v8f acc1;
    acc1[0] = lo ? 0.f : NEG;
    acc1[1] = lo ? 0.f : NEG;
    acc1[2] = lo ? 0.f : NEG;
    acc1[3] = lo ? 0.f : NEG;
    acc1[4] = NEG; acc1[5] = NEG; acc1[6] = NEG; acc1[7] = NEG;

    // ---- K loop: D=2048 in steps of 4 (512 WMMA pairs) ----
    #pragma unroll 8
    for (int k = 0; k < DDIM; k += 4) {
        const v2f a0 = *reinterpret_cast<const v2f*>(lA0 + k);
        const v2f a1 = *reinterpret_cast<const v2f*>(lA1 + k);
        const v2f bb = *reinterpret_cast<const v2f*>(gB + k);
        acc0 = __builtin_amdgcn_wmma_f32_16x16x4_f32(
            false, a0, false, bb, (short)0, acc0, false, false);
        acc1 = __builtin_amdgcn_wmma_f32_16x16x4_f32(
            false, a1, false, bb, (short)0, acc1, false, false);
    }

    // ---- Max over regions ----
    // lane l<16  : rows 0-7 (tile0) + rows 16-19 (tile1)
    // lane l+16  : rows 8-15 (tile0) + (-inf padding)
    float m = acc0[0];
    m = fmaxf(m, acc0[1]); m = fmaxf(m, acc0[2]); m = fmaxf(m, acc0[3]);
    m = fmaxf(m, acc0[4]); m = fmaxf(m, acc0[5]); m = fmaxf(m, acc0[6]);
    m = fmaxf(m, acc0[7]);
    m = fmaxf(m, acc1[0]); m = fmaxf(m, acc1[1]); m = fmaxf(m, acc1[2]);
    m = fmaxf(m, acc1[3]); m = fmaxf(m, acc1[4]); m = fmaxf(m, acc1[5]);
    m = fmaxf(m, acc1[6]); m = fmaxf(m, acc1[7]);

    // Merge the two half-wave views of the same column (wave32 shuffle).
    const float other = __shfl_xor(m, 16, 32);
    m = fmaxf(m, other);

    if (lane < 16)
        out[(size_t)b * LSEQ + col_base + mrow] = m;
}

extern "C" void kernel_launch(void* const* d_in, const int* in_sizes, int n_in,
                              void* d_out, int out_size, void* d_ws, size_t ws_size,
                              hipStream_t stream) {
    (void)in_sizes; (void)n_in; (void)d_ws; (void)ws_size; (void)out_size;
    const float* regions = (const float*)d_in[0];   // (B*20, D) fp32
    const float* words   = (const float*)d_in[1];   // (B, L, D) fp32
    float* out = (float*)d_out;                     // (B, 1, L) fp32

    const dim3 grid(BATCH * CHUNKS);  // 64 batches x 4 column-chunks of 256
    const dim3 block(THREADS);        // 16 wave32
    const size_t ldsBytes = (size_t)32 * RS * sizeof(float);  // 262656 B

    score_max_wmma_kernel<<<grid, block, ldsBytes, stream>>>(regions, words, out);
}